// DilateAttention3D_67473936220609
// MI455X (gfx1250) — compile-verified
//
#include <hip/hip_runtime.h>

// 3D dilated neighborhood attention (KS=3, DIL=1, HEAD_DIM=16) for MI455X/gfx1250.
// B=2, d=96 (6 heads x 16), D=16, H=32, W=32. fp32 in/out.
// One wave32 = one full W row (32 queries) of one (batch, head, z, y).
// QK^T on the matrix pipe via V_WMMA_F32_16X16X4_F32 chains; online softmax + AV in VALU.
// All global loads are branchless (clamped addresses + post-load select) so the
// scheduler can batch them and overlap with the WMMA chain.

typedef float v2f __attribute__((ext_vector_type(2)));
typedef float v8f __attribute__((ext_vector_type(8)));

#define CSTR 16384      // channel stride = 16*32*32

static __device__ __forceinline__ int clampi(int x, int lo, int hi) {
    return x < lo ? lo : (x > hi ? hi : x);
}

__global__ __launch_bounds__(256)
void na3d_wmma_kernel(const float* __restrict__ q,
                      const float* __restrict__ k,
                      const float* __restrict__ v,
                      float* __restrict__ out)
{
    // per-wave score staging: 2 tiles x 2 windows x 16x16 fp32 = 4KB/wave
    __shared__ float sc[8][1024];

    const int lane = threadIdx.x & 31;
    const int wv   = threadIdx.x >> 5;

    // global row id over (b, head, z, y): 2*6*16*32 = 6144 rows
    const int row = blockIdx.x * 8 + wv;
    const int y   = row & 31;
    const int z   = (row >> 5) & 15;
    const int hd  = (row >> 9) % 6;
    const int b   = row / 3072;

    const size_t chan0 = (size_t)(b * 96 + hd * 16) * CSTR;
    const float* qb = q + chan0;
    const float* kb = k + chan0;
    const float* vb = v + chan0;

    const int rowbase = (z * 32 + y) * 32;

    const int m    = lane & 15;   // query index inside tile; also B-matrix column N
    const int half = lane >> 4;
    const int Tq   = half;        // this lane's query tile (0 or 1)
    const int qw   = lane;        // this lane's query w-position (0..31)

    // ---- Q as WMMA A fragments (16x4 fp32 layout: lanes0-15 hold K=c0,c0+1;
    //      lanes16-31 hold K=c0+2,c0+3; M = lane&15), chunk t covers c=4t..4t+3
    v2f qa[2][4];
#pragma unroll
    for (int T = 0; T < 2; ++T) {
#pragma unroll
        for (int t = 0; t < 4; ++t) {
            const int c0 = 4 * t + 2 * half;
            const int w  = T * 16 + m;
            qa[T][t].x = qb[(size_t)c0 * CSTR + rowbase + w];
            qa[T][t].y = qb[(size_t)(c0 + 1) * CSTR + rowbase + w];
        }
    }

    // online-softmax state per query (per lane)
    float run_max = -1e30f;
    float ssum = 0.f;
    float o[16];
#pragma unroll
    for (int c = 0; c < 16; ++c) o[c] = 0.f;

    for (int ij = 0; ij < 9; ++ij) {
        const int di = ij / 3 - 1;
        const int dj = ij % 3 - 1;
        const int zz = z + di;
        const int yy = y + dj;
        const bool plane_ok = (zz >= 0) && (zz < 16) && (yy >= 0) && (yy < 32);
        // clamped plane base: always a legal address; invalid lanes zeroed after load
        const int pbase = (clampi(zz, 0, 15) * 32 + clampi(yy, 0, 31)) * 32;

        // ---- QK^T: per tile T, two 16-col key windows (start 16T-1 and 16T+1),
        //      each a 16x16x16 GEMM = 4 chained f32 WMMAs. OOB keys -> 0.
#pragma unroll
        for (int T = 0; T < 2; ++T) {
#pragma unroll
            for (int s = 0; s < 2; ++s) {
                const int ws   = T * 16 - 1 + 2 * s;
                const int pos  = ws + m;                 // key w for this lane's B column
                const bool okp = plane_ok && (pos >= 0) && (pos < 32);
                const int cpos = clampi(pos, 0, 31);

                // branchless loads of all 4 B chunks first (batched, one wait)
                v2f bm[4];
#pragma unroll
                for (int t = 0; t < 4; ++t) {
                    const int c0 = 4 * t + 2 * half;
                    const float bx = kb[(size_t)c0 * CSTR + pbase + cpos];
                    const float by = kb[(size_t)(c0 + 1) * CSTR + pbase + cpos];
                    bm[t].x = okp ? bx : 0.f;            // v_cndmask, no branch
                    bm[t].y = okp ? by : 0.f;
                }

                v8f acc = {};
#pragma unroll
                for (int t = 0; t < 4; ++t)
                    acc = __builtin_amdgcn_wmma_f32_16x16x4_f32(
                              false, qa[T][t], false, bm[t], (short)0, acc, false, false);

                // D layout: lane holds N=m fixed, M = 8*half + r -> contiguous LDS run
                float* dst = &sc[wv][(T * 2 + s) * 256 + m * 16 + half * 8];
#pragma unroll
                for (int r = 0; r < 8; ++r) dst[r] = acc[r];
            }
        }

        // ---- gather this query's 3 scores (key w = qw + lo - 1)
        float s3[3];
#pragma unroll
        for (int lo = 0; lo < 3; ++lo) {
            const int n1 = m + lo;           // column in window0 = m + (lo-1) + 1
            float val;
            if (n1 <= 15)
                val = sc[wv][(Tq * 2 + 0) * 256 + n1 * 16 + m];
            else
                val = sc[wv][(Tq * 2 + 1) * 256 + (n1 - 2) * 16 + m];
            s3[lo] = val * 0.25f;            // scale = HEAD_DIM^-0.5
        }

        // ---- online softmax + AV accumulation (branchless V loads)
        const float nm = fmaxf(run_max, fmaxf(fmaxf(s3[0], s3[1]), s3[2]));
        const float alpha = __expf(run_max - nm);
        ssum *= alpha;
#pragma unroll
        for (int c = 0; c < 16; ++c) o[c] *= alpha;
#pragma unroll
        for (int lo = 0; lo < 3; ++lo) {
            const float p = __expf(s3[lo] - nm);
            ssum += p;                       // zero-padded neighbors still count (score 0)
            const int key  = qw + lo - 1;
            const bool okv = plane_ok && (key >= 0) && (key < 32);
            const int ckey = clampi(key, 0, 31);
            const float pm = okv ? p : 0.f;  // mask the weight, not the load
            const float* vp = vb + pbase + ckey;
#pragma unroll
            for (int c = 0; c < 16; ++c)
                o[c] = fmaf(pm, vp[(size_t)c * CSTR], o[c]);
        }
        run_max = nm;
    }

    // ---- out[b, z, y, w, hd*16 + c]
    const float inv = 1.0f / ssum;
    float* op = out + (((size_t)((b * 16 + z) * 32 + y) * 32 + qw) * 96 + hd * 16);
#pragma unroll
    for (int c = 0; c < 16; ++c) op[c] = o[c] * inv;
}

extern "C" void kernel_launch(void* const* d_in, const int* in_sizes, int n_in,
                              void* d_out, int out_size, void* d_ws, size_t ws_size,
                              hipStream_t stream) {
    (void)in_sizes; (void)n_in; (void)out_size; (void)d_ws; (void)ws_size;
    const float* q = (const float*)d_in[0];
    const float* k = (const float*)d_in[1];
    const float* v = (const float*)d_in[2];
    float* out = (float*)d_out;

    // 6144 (b,head,z,y) rows, 8 waves (rows) per 256-thread block
    dim3 grid(768), block(256);
    hipLaunchKernelGGL(na3d_wmma_kernel, grid, block, 0, stream, q, k, v, out);
}